// QuaternionTorusBrain_24352464570154
// MI455X (gfx1250) — compile-verified
//
#include <hip/hip_runtime.h>
#include <hip/hip_bf16.h>

// ---------------------------------------------------------------------------
// Quaternion torus spectral conv for MI455X (gfx1250, wave32, WMMA + TDM).
// Pipeline: rfft2 -> bin-major transposes -> per-bin quaternion complex GEMM
// on v_wmma_f32_16x16x4_f32 with tensor_load_to_lds staging -> transpose back
// -> irfft2.
// ---------------------------------------------------------------------------

#define GH 64
#define GW 64
#define FW 33                 // GW/2+1
#define NBIN (GH * FW)        // 2112
#define NQ 64
#define NB 32                 // batch
#define NIMG (NB * 4 * NQ)    // 8192 images of 64x64
#define PLANE ((size_t)NIMG * NBIN)   // 17,301,504 floats per re/im plane

// LDS pitches in qgemm: 64 data floats + 4 pad floats = 68 (272B = 16*17,
// keeps 16B alignment for b128/TDM writes and kills stride-64 bank conflicts).
#define QPITCH 68
#define XS_ROWS 256           // 2 parts * 4 comps * 32 batch
#define KS_ROWS 512           // 2 parts * 4 wc * 64 i
#define XS_FLOATS (XS_ROWS * QPITCH)          // 17408
#define KS_FLOATS (KS_ROWS * QPITCH)          // 34816
#define QGEMM_LDS_FLOATS (XS_FLOATS + KS_FLOATS)   // 52224
#define QGEMM_LDS_BYTES (QGEMM_LDS_FLOATS * 4)     // 208896

typedef float v2f __attribute__((ext_vector_type(2)));
typedef float v8f __attribute__((ext_vector_type(8)));
typedef unsigned int u32x4 __attribute__((ext_vector_type(4)));
typedef int i32x4 __attribute__((ext_vector_type(4)));
typedef int i32x8 __attribute__((ext_vector_type(8)));

__device__ __forceinline__ int brev6(int x) { return (int)(__brev((unsigned)x) >> 26); }

// ---------------------------------------------------------------------------
// Tensor Data Mover: 1-D tile load global->LDS with hardware row padding.
// D# packed per CDNA5 ISA:
//   group0: count=1 | lds_addr | global_addr | type=2
//   group1: data_size=4B, pad_enable, pad_interval code, pad_amount code,
//           tensor_dim0 = tile_dim0 = nelem (1-D: tile_dim1 = 0)
// pad codes: interval 5 -> every 64 DWORDs, amount 3 -> 4 DWORDs  (pitch 68)
// This toolchain's builtin is the 6-arg form:
//   (u32x4 g0, i32x8 g1, i32x4 g2, i32x4 g3, i32x8 extra, i32 cpol)
// ---------------------------------------------------------------------------
#if __has_builtin(__builtin_amdgcn_tensor_load_to_lds)
#define HAVE_TDM 1
__device__ __forceinline__ void tdm_load_1d_padded(unsigned int lds_byte_addr,
                                                   const void* gptr,
                                                   unsigned int nelem) {
  unsigned long long ga = (unsigned long long)(uintptr_t)gptr;
  u32x4 g0;
  g0[0] = 1u;                                   // count=1, user-mode
  g0[1] = lds_byte_addr;                        // lds_addr
  g0[2] = (unsigned int)ga;                     // global_addr[31:0]
  g0[3] = ((unsigned int)(ga >> 32) & 0x1FFFFFFu) | (2u << 30);  // addr[56:32] | type=2
  i32x8 g1;
  g1[0] = (int)((2u << 16)        // data_size = 4 bytes
              | (1u << 20)        // pad_enable
              | (5u << 22)        // pad_interval: 64 DWORDs
              | (3u << 25));      // pad_amount:   4 DWORDs
  g1[1] = (int)((nelem & 0xFFFFu) << 16);       // tensor_dim0[15:0] @ bits 63:48
  g1[2] = (int)(nelem >> 16);                   // tensor_dim0[31:16]; tensor_dim1=0
  g1[3] = (int)((nelem & 0xFFFFu) << 16);       // tile_dim0 @ bits 127:112
  g1[4] = 0;                                    // tile_dim1=0, tile_dim2=0 (1-D)
  g1[5] = (int)nelem;                           // tensor_dim0_stride[31:0]
  g1[6] = 0;
  g1[7] = 0;
  i32x4 z4 = {0, 0, 0, 0};
  i32x8 z8 = {0, 0, 0, 0, 0, 0, 0, 0};
  __builtin_amdgcn_tensor_load_to_lds(g0, g1, z4, z4, z8, 0);
}
#else
#define HAVE_TDM 0
#endif

// ---------------------------------------------------------------------------
// In-LDS 64-point radix-2 FFTs. Data lives in re[64*65], im[64*65] (padded
// rows). blockDim.x == 256 assumed. dirsign = -1 fwd, +1 inverse (unnorm.)
// ---------------------------------------------------------------------------
__device__ __forceinline__ void fft_rows(float* re, float* im, float dirsign) {
  const int tid = threadIdx.x;
  for (int k = tid; k < 64 * 64; k += 256) {
    int r = k >> 6, j = k & 63;
    int rj = brev6(j);
    if (j < rj) {
      int a = r * 65 + j, b = r * 65 + rj;
      float t0 = re[a]; re[a] = re[b]; re[b] = t0;
      float t1 = im[a]; im[a] = im[b]; im[b] = t1;
    }
  }
  __syncthreads();
  for (int s = 1; s <= 6; ++s) {
    const int m = 1 << s, half = m >> 1;
    const float ang0 = dirsign * 6.28318530717958647692f / (float)m;
    for (int k = tid; k < 64 * 32; k += 256) {
      int r = k >> 5, j = k & 31;
      int blk = j >> (s - 1), t = j & (half - 1);
      int i0 = r * 65 + blk * m + t;
      int i1 = i0 + half;
      float wi, wr;
      __sincosf(ang0 * (float)t, &wi, &wr);
      float br = re[i1], bi = im[i1];
      float vr = br * wr - bi * wi;
      float vi = br * wi + bi * wr;
      float ur = re[i0], ui = im[i0];
      re[i0] = ur + vr; im[i0] = ui + vi;
      re[i1] = ur - vr; im[i1] = ui - vi;
    }
    __syncthreads();
  }
}

__device__ __forceinline__ void fft_cols(float* re, float* im, int ncols, float dirsign) {
  const int tid = threadIdx.x;
  for (int k = tid; k < ncols * 64; k += 256) {
    int c = k >> 6, j = k & 63;
    int rj = brev6(j);
    if (c < ncols && j < rj) {
      int a = j * 65 + c, b = rj * 65 + c;
      float t0 = re[a]; re[a] = re[b]; re[b] = t0;
      float t1 = im[a]; im[a] = im[b]; im[b] = t1;
    }
  }
  __syncthreads();
  for (int s = 1; s <= 6; ++s) {
    const int m = 1 << s, half = m >> 1;
    const float ang0 = dirsign * 6.28318530717958647692f / (float)m;
    for (int k = tid; k < ncols * 32; k += 256) {
      int c = k >> 5, j = k & 31;
      int blk = j >> (s - 1), t = j & (half - 1);
      int i0 = (blk * m + t) * 65 + c;
      int i1 = i0 + half * 65;
      float wi, wr;
      __sincosf(ang0 * (float)t, &wi, &wr);
      float br = re[i1], bi = im[i1];
      float vr = br * wr - bi * wi;
      float vi = br * wi + bi * wr;
      float ur = re[i0], ui = im[i0];
      re[i0] = ur + vr; im[i0] = ui + vi;
      re[i1] = ur - vr; im[i1] = ui - vi;
    }
    __syncthreads();
  }
}

// ---------------------------------------------------------------------------
// Kernel 1: forward rfft2. One block per image. Output image-major planes.
// ---------------------------------------------------------------------------
__global__ __launch_bounds__(256) void fft_fwd(const float* __restrict__ x,
                                               float* __restrict__ Xre,
                                               float* __restrict__ Xim) {
  __shared__ float re[64 * 65];
  __shared__ float im[64 * 65];
  const int img = blockIdx.x;
  const int tid = threadIdx.x;
  const float4* src = (const float4*)(x + (size_t)img * 4096);
  for (int k = tid; k < 1024; k += 256) {
    int r = k >> 4, c = (k & 15) * 4;
    float4 v = src[k];
    float* dr = &re[r * 65 + c];
    float* di = &im[r * 65 + c];
    dr[0] = v.x; dr[1] = v.y; dr[2] = v.z; dr[3] = v.w;
    di[0] = 0.f; di[1] = 0.f; di[2] = 0.f; di[3] = 0.f;
  }
  __syncthreads();
  fft_rows(re, im, -1.f);
  fft_cols(re, im, FW, -1.f);
  float* outre = Xre + (size_t)img * NBIN;
  float* outim = Xim + (size_t)img * NBIN;
  for (int k = tid; k < NBIN; k += 256) {
    int h = k / FW, wf = k % FW;
    outre[k] = re[h * 65 + wf];
    outim[k] = im[h * 65 + wf];
  }
}

// ---------------------------------------------------------------------------
// Kernel 2: transpose X to bin-major: out[bin][c*2048 + b*64 + q].
// in: [img][bin], img = b*256 + c*64 + q.  grid (66, 256, 2).
// ---------------------------------------------------------------------------
__global__ __launch_bounds__(256) void transpose_x(const float* __restrict__ in,
                                                   float* __restrict__ out) {
  __shared__ float tile[32][33];
  const float* src = in + (size_t)blockIdx.z * PLANE;
  float* dst = out + (size_t)blockIdx.z * PLANE;
  const int bin0 = blockIdx.x * 32, img0 = blockIdx.y * 32;
  const int tx = threadIdx.x & 31, ty = threadIdx.x >> 5;
  for (int r = ty; r < 32; r += 8)
    tile[r][tx] = src[(size_t)(img0 + r) * NBIN + bin0 + tx];
  __syncthreads();
  const int b = img0 >> 8, c = (img0 >> 6) & 3, q0 = img0 & 63;
  const size_t obase = (size_t)c * 2048 + (size_t)b * 64 + q0;
  for (int r = ty; r < 32; r += 8)
    dst[(size_t)(bin0 + r) * 8192 + obase + tx] = tile[tx][r];
}

// ---------------------------------------------------------------------------
// Kernel 3: transpose 8 kernel tensors [i][o][bin] to bin-major blocks
// Kb[bin*32768 + (part*4+wc)*4096 + i*64 + o].  grid (66, 128, 8).
// ---------------------------------------------------------------------------
__global__ __launch_bounds__(256) void transpose_k(
    const float* __restrict__ p0, const float* __restrict__ p1,
    const float* __restrict__ p2, const float* __restrict__ p3,
    const float* __restrict__ p4, const float* __restrict__ p5,
    const float* __restrict__ p6, const float* __restrict__ p7,
    float* __restrict__ kb) {
  __shared__ float tile[32][33];
  const float* srcs[8] = {p0, p1, p2, p3, p4, p5, p6, p7};
  const int t = blockIdx.z;
  const float* src = srcs[t];
  const int part = t & 1, wc = t >> 1;   // order: kr_w, ki_w, kr_x, ki_x, ...
  const int bin0 = blockIdx.x * 32, io0 = blockIdx.y * 32;
  const int tx = threadIdx.x & 31, ty = threadIdx.x >> 5;
  for (int r = ty; r < 32; r += 8)
    tile[r][tx] = src[(size_t)(io0 + r) * NBIN + bin0 + tx];
  __syncthreads();
  const size_t sub = (size_t)(part * 4 + wc) * 4096 + io0;
  for (int r = ty; r < 32; r += 8)
    kb[(size_t)(bin0 + r) * 32768 + sub + tx] = tile[tx][r];
}

// ---------------------------------------------------------------------------
// Kernel 4: per-bin quaternion complex GEMM on WMMA f32 16x16x4.
// One WG (8 waves) per bin. LDS staged by the Tensor Data Mover (wave 0
// issues 3 DMAs with hardware 64->68 DWORD row padding), waves compute
// 1024 v_wmma each on 2x4 16x16 accumulator tiles.
// ---------------------------------------------------------------------------
__global__ __launch_bounds__(256) void qgemm(const float* __restrict__ Xre,
                                             const float* __restrict__ Xim,
                                             const float* __restrict__ Kb,
                                             float* __restrict__ Pre,
                                             float* __restrict__ Pim) {
  extern __shared__ float lds[];
  float* Xs = lds;                 // [2][4][32] rows x QPITCH
  float* Ks = lds + XS_FLOATS;     // [2][4][64] rows x QPITCH
  const int bin = blockIdx.x;
  const int tid = threadIdx.x;

#if HAVE_TDM
  if (tid < 32) {
    const unsigned int xs_base = (unsigned int)(uintptr_t)&Xs[0];
    const unsigned int ks_base = (unsigned int)(uintptr_t)&Ks[0];
    // X_re: 8192 elems (128 rows of 64), X_im after the padded X_re block.
    tdm_load_1d_padded(xs_base, Xre + (size_t)bin * 8192, 8192);
    tdm_load_1d_padded(xs_base + 128u * QPITCH * 4u, Xim + (size_t)bin * 8192, 8192);
    // K: 32768 elems (512 rows of 64).
    tdm_load_1d_padded(ks_base, Kb + (size_t)bin * 32768, 32768);
    __builtin_amdgcn_s_wait_tensorcnt(0);
  }
#else
  {
    const float4* xr = (const float4*)(Xre + (size_t)bin * 8192);
    const float4* xi = (const float4*)(Xim + (size_t)bin * 8192);
    for (int k = tid; k < 2048; k += 256) {
      int row = k >> 4, c = (k & 15) * 4;
      float4 vr = xr[k];
      float4 vi = xi[k];
      float* dr = &Xs[row * QPITCH + c];
      float* di = &Xs[(128 + row) * QPITCH + c];
      dr[0] = vr.x; dr[1] = vr.y; dr[2] = vr.z; dr[3] = vr.w;
      di[0] = vi.x; di[1] = vi.y; di[2] = vi.z; di[3] = vi.w;
    }
    const float4* kk = (const float4*)(Kb + (size_t)bin * 32768);
    for (int k = tid; k < 8192; k += 256) {
      int row = k >> 4, c = (k & 15) * 4;
      float4 v = kk[k];
      float* d = &Ks[row * QPITCH + c];
      d[0] = v.x; d[1] = v.y; d[2] = v.z; d[3] = v.w;
    }
  }
#endif
  __syncthreads();

  const int wave = tid >> 5, lane = tid & 31;
  const int cout = wave >> 1, pp = wave & 1;     // pp: 0 = real, 1 = imag
  const int lrow = lane & 15;
  const int khi = (lane >> 4) << 1;              // lanes 16-31 -> K-slot +2

  // Hamilton tables: P[c] = sum_xc sg[c][xc] * C[wc[c][xc], xc]
  const int wc_tab[4][4] = {{0, 1, 2, 3}, {1, 0, 3, 2}, {2, 3, 0, 1}, {3, 2, 1, 0}};
  const float sg_tab[4][4] = {{1.f, -1.f, -1.f, -1.f},
                              {1.f, 1.f, -1.f, 1.f},
                              {1.f, 1.f, 1.f, -1.f},
                              {1.f, -1.f, 1.f, 1.f}};

  v8f acc[2][4];
#pragma unroll
  for (int mt = 0; mt < 2; ++mt)
#pragma unroll
    for (int nt = 0; nt < 4; ++nt)
      acc[mt][nt] = (v8f){0.f, 0.f, 0.f, 0.f, 0.f, 0.f, 0.f, 0.f};

  for (int xc = 0; xc < 4; ++xc) {
    const int wc = wc_tab[cout][xc];
    const float s = sg_tab[cout][xc];
#pragma unroll
    for (int sub = 0; sub < 2; ++sub) {
      // real out: +Xre*Kre - Xim*Kim ; imag out: +Xre*Kim + Xim*Kre
      const int ap = sub;
      const int bp = pp ^ sub;
      const float sf = (pp == 0 && sub == 1) ? -s : s;
      const float* xbase = &Xs[(ap * 128 + xc * 32) * QPITCH];
      const float* kbase = &Ks[(bp * 4 + wc) * 64 * QPITCH];
#pragma unroll
      for (int kc = 0; kc < 16; ++kc) {
        const int i0 = kc * 4 + khi;
        v2f a0, a1, bfrag[4];
        // A (X) fragments; Hamilton/complex sign folded in via VALU
        // (f32 WMMA exposes C-negate only).
        a0.x = xbase[lrow * QPITCH + i0] * sf;
        a0.y = xbase[lrow * QPITCH + i0 + 1] * sf;
        a1.x = xbase[(16 + lrow) * QPITCH + i0] * sf;
        a1.y = xbase[(16 + lrow) * QPITCH + i0 + 1] * sf;
#pragma unroll
        for (int nt = 0; nt < 4; ++nt) {
          bfrag[nt].x = kbase[i0 * QPITCH + nt * 16 + lrow];
          bfrag[nt].y = kbase[(i0 + 1) * QPITCH + nt * 16 + lrow];
        }
#pragma unroll
        for (int nt = 0; nt < 4; ++nt) {
          acc[0][nt] = __builtin_amdgcn_wmma_f32_16x16x4_f32(
              false, a0, false, bfrag[nt], (short)0, acc[0][nt], false, false);
          acc[1][nt] = __builtin_amdgcn_wmma_f32_16x16x4_f32(
              false, a1, false, bfrag[nt], (short)0, acc[1][nt], false, false);
        }
      }
    }
  }

  // ---- store P bin-major: plane[pp][bin][cout*2048 + b*64 + o] ----
  float* outp = (pp ? Pim : Pre) + (size_t)bin * 8192 + (size_t)cout * 2048;
  const int rbase = (lane >> 4) << 3;   // lanes 16-31 hold rows +8
#pragma unroll
  for (int mt = 0; mt < 2; ++mt)
#pragma unroll
    for (int nt = 0; nt < 4; ++nt)
#pragma unroll
      for (int v = 0; v < 8; ++v) {
        int brow = mt * 16 + rbase + v;
        int o = nt * 16 + lrow;
        outp[brow * 64 + o] = acc[mt][nt][v];
      }
}

// ---------------------------------------------------------------------------
// Kernel 5: transpose P back to image-major: out[img2][bin],
// img2 = b*256 + c*64 + o from pos = c*2048 + b*64 + o.  grid (66, 256, 2).
// ---------------------------------------------------------------------------
__global__ __launch_bounds__(256) void transpose_p(const float* __restrict__ in,
                                                   float* __restrict__ out) {
  __shared__ float tile[32][33];
  const float* src = in + (size_t)blockIdx.z * PLANE;
  float* dst = out + (size_t)blockIdx.z * PLANE;
  const int bin0 = blockIdx.x * 32, pos0 = blockIdx.y * 32;
  const int tx = threadIdx.x & 31, ty = threadIdx.x >> 5;
  for (int r = ty; r < 32; r += 8)
    tile[r][tx] = src[(size_t)(bin0 + r) * 8192 + pos0 + tx];
  __syncthreads();
  const int c = pos0 >> 11, b = (pos0 >> 6) & 31, o0 = pos0 & 63;
  for (int r = ty; r < 32; r += 8) {
    size_t img2 = (size_t)b * 256 + (size_t)c * 64 + o0 + r;
    dst[img2 * NBIN + bin0 + tx] = tile[tx][r];
  }
}

// ---------------------------------------------------------------------------
// Kernel 6: irfft2. Inverse column FFT on 33 columns, Hermitian fill,
// inverse row FFT, scale 1/4096.
// ---------------------------------------------------------------------------
__global__ __launch_bounds__(256) void fft_inv(const float* __restrict__ Pre,
                                               const float* __restrict__ Pim,
                                               float* __restrict__ out) {
  __shared__ float re[64 * 65];
  __shared__ float im[64 * 65];
  const int img = blockIdx.x;
  const int tid = threadIdx.x;
  const float* sre = Pre + (size_t)img * NBIN;
  const float* sim = Pim + (size_t)img * NBIN;
  for (int k = tid; k < NBIN; k += 256) {
    int h = k / FW, wf = k % FW;
    re[h * 65 + wf] = sre[k];
    im[h * 65 + wf] = sim[k];
  }
  __syncthreads();
  fft_cols(re, im, FW, +1.f);
  for (int k = tid; k < 64 * 31; k += 256) {
    int r = k / 31, w = 33 + (k % 31);
    re[r * 65 + w] = re[r * 65 + 64 - w];
    im[r * 65 + w] = -im[r * 65 + 64 - w];
  }
  __syncthreads();
  fft_rows(re, im, +1.f);
  float* dst = out + (size_t)img * 4096;
  const float scale = 1.f / 4096.f;
  for (int k = tid; k < 4096; k += 256) {
    int r = k >> 6, c = k & 63;
    dst[k] = re[r * 65 + c] * scale;
  }
}

// ---------------------------------------------------------------------------
// Host-side launcher.
// ws layout (floats): A_re | A_im | B_re | B_im | Kb(4*PLANE) == 8*PLANE.
// A is reused for Pb after transpose_x, B is reused for Pim after qgemm.
// ---------------------------------------------------------------------------
extern "C" void kernel_launch(void* const* d_in, const int* in_sizes, int n_in,
                              void* d_out, int out_size, void* d_ws, size_t ws_size,
                              hipStream_t stream) {
  (void)in_sizes; (void)n_in; (void)out_size; (void)ws_size;
  const float* x = (const float*)d_in[0];
  const float* kp[8];
  for (int i = 0; i < 8; ++i) kp[i] = (const float*)d_in[1 + i];

  float* ws = (float*)d_ws;
  float* Are = ws;
  float* Aim = ws + PLANE;
  float* Bre = ws + 2 * PLANE;
  float* Bim = ws + 3 * PLANE;
  float* Kb  = ws + 4 * PLANE;
  float* out = (float*)d_out;

  fft_fwd<<<NIMG, 256, 0, stream>>>(x, Are, Aim);
  transpose_x<<<dim3(66, 256, 2), 256, 0, stream>>>(Are, Bre);
  transpose_k<<<dim3(66, 128, 8), 256, 0, stream>>>(kp[0], kp[1], kp[2], kp[3],
                                                    kp[4], kp[5], kp[6], kp[7], Kb);
  qgemm<<<NBIN, 256, QGEMM_LDS_BYTES, stream>>>(Bre, Bim, Kb, Are, Aim);
  transpose_p<<<dim3(66, 256, 2), 256, 0, stream>>>(Are, Bre);
  fft_inv<<<NIMG, 256, 0, stream>>>(Bre, Bim, out);
}